// TemporalModule_4183298146525
// MI455X (gfx1250) — compile-verified
//
#include <hip/hip_runtime.h>
#include <hip/hip_bf16.h>
#include <math.h>

// ---------------------------------------------------------------------------
// Problem constants (reference setup_inputs)
// ---------------------------------------------------------------------------
#define B   64
#define S   30
#define C   62
#define L   169
#define GCN 64
#define GRU 64
#define L1  106   // 169 - 64 + 1
#define L2  75    // 106 - 32 + 1
#define L3  62    // 75  - 14 + 1

// padded dims (all WMMA operands unguarded)
#define CP   64    // channel pad (M and K of pointwise GEMMs)
#define LP   172   // L padded to mult of 4  (K of gcn GEMM)
#define LP2  176   // L padded to mult of 16 (N and K of DFT GEMM)
#define L1P  112   // L1 -> mult of 16
#define L2P  80    // L2 -> mult of 16

typedef float v2f __attribute__((ext_vector_type(2)));
typedef float v8f __attribute__((ext_vector_type(8)));
typedef unsigned int u32x4 __attribute__((ext_vector_type(4)));
typedef int i32x4 __attribute__((ext_vector_type(4)));
typedef int i32x8 __attribute__((ext_vector_type(8)));

#define V8F_ZERO {0.f,0.f,0.f,0.f,0.f,0.f,0.f,0.f}

#if defined(__has_builtin)
#  if __has_builtin(__builtin_amdgcn_tensor_load_to_lds) && \
      __has_builtin(__builtin_amdgcn_s_wait_tensorcnt)
#    define HAVE_TDM 1
#  endif
#endif

__device__ __forceinline__ float sigmoidf(float x) { return 1.f / (1.f + expf(-x)); }

// ---------------------------------------------------------------------------
// Stage A: squeeze-and-excitation
// ---------------------------------------------------------------------------
__global__ void se_mean_k(const float* __restrict__ x, float* __restrict__ sq) {
  __shared__ float red[256];
  int bs = blockIdx.x;                        // b*S + s
  const float* p = x + (size_t)bs * C * L;
  float acc = 0.f;
  for (int i = threadIdx.x; i < C * L; i += 256) acc += p[i];
  red[threadIdx.x] = acc;
  __syncthreads();
  for (int s = 128; s > 0; s >>= 1) {
    if ((int)threadIdx.x < s) red[threadIdx.x] += red[threadIdx.x + s];
    __syncthreads();
  }
  if (threadIdx.x == 0) sq[bs] = red[0] / (float)(C * L);
}

__global__ void se_mlp_k(const float* __restrict__ sq, const float* __restrict__ w1,
                         const float* __restrict__ w2, float* __restrict__ a) {
  int b = blockIdx.x;
  __shared__ float hid[S / 2];
  int t = (int)threadIdx.x;
  if (t < S / 2) {
    float acc = 0.f;
    for (int s = 0; s < S; ++s) acc += sq[b * S + s] * w1[s * (S / 2) + t];
    hid[t] = fmaxf(acc, 0.f);
  }
  __syncthreads();
  if (t < S) {
    float acc = 0.f;
    for (int j = 0; j < S / 2; ++j) acc += hid[j] * w2[j * S + t];
    a[b * S + t] = sigmoidf(acc);
  }
}

// writes into zero-padded (B x CP x LP2) layout
__global__ void max_seg_k(const float* __restrict__ x, const float* __restrict__ a,
                          float* __restrict__ ms) {
  int idx = blockIdx.x * blockDim.x + threadIdx.x;
  if (idx >= B * C * L) return;
  int l = idx % L;
  int c = (idx / L) % C;
  int b = idx / (L * C);
  float m = -INFINITY;
  for (int s = 0; s < S; ++s) {
    float v = x[(((size_t)b * S + s) * C + c) * L + l] * a[b * S + s];
    m = fmaxf(m, v);
  }
  ms[((size_t)b * CP + c) * LP2 + l] = m;
}

// ---------------------------------------------------------------------------
// Stage B: iCOH (DFT as two fp32 WMMA GEMMs; only adjacency row C-1 consumed)
// ---------------------------------------------------------------------------
__global__ void make_dft_k(float* __restrict__ ct, float* __restrict__ st) {
  int idx = blockIdx.x * blockDim.x + threadIdx.x;
  if (idx >= LP2 * LP2) return;
  int f = idx % LP2, l = idx / LP2;
  float cv = 0.f, sv = 0.f;
  if (f < L && l < L) {
    int m = (l * f) % L;                       // exact phase reduction
    float th = -2.f * 3.14159265358979323846f * (float)m / (float)L;
    sincosf(th, &sv, &cv);
  }
  ct[idx] = cv;                                // Fre = X @ ct
  st[idx] = sv;                                // Fim = X @ st (sign baked in)
}

__global__ void dft_wmma_k(const float* __restrict__ ms, const float* __restrict__ ct,
                           const float* __restrict__ st, float* __restrict__ fre,
                           float* __restrict__ fim, float* __restrict__ fab) {
  int b = blockIdx.z;
  int n0 = blockIdx.x * 16, m0 = blockIdx.y * 16;
  const float* A = ms + (size_t)b * CP * LP2;
  int lane = (int)(threadIdx.x & 31);
  int arow = m0 + (lane & 15);
  int col = n0 + (lane & 15);
  v8f aR = V8F_ZERO, aI = V8F_ZERO;
  for (int k0 = 0; k0 < LP2; k0 += 4) {
    int k = k0 + ((lane >> 4) << 1);
    v2f a = *(const v2f*)(A + (size_t)arow * LP2 + k);
    v2f br, bi;
    br.x = ct[(size_t)k * LP2 + col];
    br.y = ct[(size_t)(k + 1) * LP2 + col];
    aR = __builtin_amdgcn_wmma_f32_16x16x4_f32(false, a, false, br, (short)0, aR, false, false);
    bi.x = st[(size_t)k * LP2 + col];
    bi.y = st[(size_t)(k + 1) * LP2 + col];
    aI = __builtin_amdgcn_wmma_f32_16x16x4_f32(false, a, false, bi, (short)0, aI, false, false);
  }
  int rbase = m0 + ((lane >> 4) << 3);
#pragma unroll
  for (int r = 0; r < 8; ++r) {
    size_t idx = ((size_t)b * CP + rbase + r) * LP2 + col;
    float re = aR[r], im = aI[r];
    fre[idx] = re;
    fim[idx] = im;
    fab[idx] = sqrtf(re * re + im * im);
  }
}

__global__ void icoh_row_k(const float* __restrict__ fre, const float* __restrict__ fim,
                           const float* __restrict__ fab, float* __restrict__ arow) {
  int b = blockIdx.x;
  int j = (int)threadIdx.x;
  if (j >= C) return;
  const float* re61 = fre + ((size_t)b * CP + (C - 1)) * LP2;
  const float* im61 = fim + ((size_t)b * CP + (C - 1)) * LP2;
  const float* ab61 = fab + ((size_t)b * CP + (C - 1)) * LP2;
  const float* rej = fre + ((size_t)b * CP + j) * LP2;
  const float* imj = fim + ((size_t)b * CP + j) * LP2;
  const float* abj = fab + ((size_t)b * CP + j) * LP2;
  float num = 0.f, den = 0.f;
  for (int f = 0; f < L; ++f) {
    num += im61[f] * rej[f] - re61[f] * imj[f];   // Im(F_61 * conj(F_j))
    den += ab61[f] * abj[f];
  }
  arow[b * C + j] = (j == C - 1) ? 0.f : (num / den);
}

// ---------------------------------------------------------------------------
// Stage C per step
// ---------------------------------------------------------------------------
__global__ void dw_conv_k(const float* __restrict__ in, size_t batchStride, int inRS,
                          const float* __restrict__ wk, const float* __restrict__ wb,
                          int Kw, float* __restrict__ out, int outRS, int Lout) {
  int ch = blockIdx.x, b = blockIdx.y;
  extern __shared__ float sw[];
  for (int i = threadIdx.x; i < Kw; i += blockDim.x) sw[i] = wk[ch * Kw + i];
  __syncthreads();
  const float* row = in + (size_t)b * batchStride + (size_t)ch * inRS;
  __builtin_prefetch(row, 0, 0);              // global_prefetch_b8
  int l = (int)threadIdx.x;
  if (l < Lout) {
    float acc = wb[ch];
#pragma unroll 4
    for (int k = 0; k < Kw; ++k) acc += row[l + k] * sw[k];
    out[((size_t)b * CP + ch) * outRS + l] = acc;
  }
}

// out[b] (CP x Np) = PW (CP x CP, padded) @ in[b] (CP x Np) + bias[row]
// Block = 4 waves = all 4 M-tiles of one N-tile; A staged in LDS (TDM if available).
__global__ void pw_gemm_k(const float* __restrict__ pwp, const float* __restrict__ biasp,
                          const float* __restrict__ in, float* __restrict__ out, int Np) {
  __shared__ float sA[CP * CP];
  int b = blockIdx.z;
  int n0 = blockIdx.x * 16;
  int wave = (int)(threadIdx.x >> 5);
  int lane = (int)(threadIdx.x & 31);
#if defined(HAVE_TDM)
  if (wave == 0) {
    // Tensor DMA: 1-D tile of 4096 4-byte elements, global -> LDS (ISA 8.3/8.4)
    unsigned long long ga = (unsigned long long)(uintptr_t)pwp;
    unsigned int lds = (unsigned int)(uintptr_t)(&sA[0]);
    const unsigned int elems = CP * CP;
    u32x4 g0;
    g0[0] = 1u;                                              // count=1, user desc
    g0[1] = lds;                                             // lds_addr
    g0[2] = (unsigned int)(ga & 0xFFFFFFFFull);              // global_addr[31:0]
    g0[3] = (unsigned int)((ga >> 32) & 0x1FFFFFFull) | (2u << 30);  // [56:32] | type=2
    i32x8 g1;
    g1[0] = (int)(2u << 16);                                 // data_size=4B, mask=0
    g1[1] = (int)((elems & 0xFFFFu) << 16);                  // tensor_dim0 lo16
    g1[2] = (int)(((elems >> 16) & 0xFFFFu) | (1u << 16));   // td0 hi16 | tensor_dim1=1
    g1[3] = (int)((elems & 0xFFFFu) << 16);                  // tile_dim0 = elems
    g1[4] = 0;                                               // tile_dim1/2 unused
    g1[5] = (int)elems;                                      // tensor_dim0_stride lo32
    g1[6] = 0;
    g1[7] = 0;
    i32x4 gz4 = {0, 0, 0, 0};
    i32x8 gz8 = {0, 0, 0, 0, 0, 0, 0, 0};
    __builtin_amdgcn_tensor_load_to_lds(g0, g1, gz4, gz4, gz8, 0);
    __builtin_amdgcn_s_wait_tensorcnt(0);
  }
  __syncthreads();
#else
  for (int i = threadIdx.x; i < CP * CP; i += blockDim.x) sA[i] = pwp[i];
  __syncthreads();
#endif
  int m0 = wave * 16;
  int arow = m0 + (lane & 15);
  int col = n0 + (lane & 15);
  const float* Bm = in + (size_t)b * CP * Np;
  v8f acc = V8F_ZERO;
  for (int k0 = 0; k0 < CP; k0 += 4) {
    int k = k0 + ((lane >> 4) << 1);
    v2f a = *(const v2f*)(sA + arow * CP + k);
    v2f bf;
    bf.x = Bm[(size_t)k * Np + col];
    bf.y = Bm[(size_t)(k + 1) * Np + col];
    acc = __builtin_amdgcn_wmma_f32_16x16x4_f32(false, a, false, bf, (short)0, acc, false, false);
  }
  float* D = out + (size_t)b * CP * Np;
  int rbase = m0 + ((lane >> 4) << 3);
#pragma unroll
  for (int r = 0; r < 8; ++r) {
    int row = rbase + r;
    D[(size_t)row * Np + col] = acc[r] + biasp[row];
  }
}

// Row-61 collapse: d61 = pw3[61,:] @ y3[b] + pwb3[61]; arow_t = sigmoid(d61 + adjrow);
// v[b,l] = arow_t . xt[b,:,l]   (vrow padded B x LP, pads pre-zeroed)
__global__ void row_stage_k(const float* __restrict__ x, int t,
                            const float* __restrict__ pwk3, const float* __restrict__ pwb3,
                            const float* __restrict__ y3, const float* __restrict__ adjrow,
                            float* __restrict__ vout) {
  int b = blockIdx.x;
  __shared__ float sA[C];
  int tid = (int)threadIdx.x;
  if (tid < C) {
    float d = pwb3[C - 1];
    const float* y3b = y3 + (size_t)b * CP * CP;
    for (int i = 0; i < C; ++i) d += pwk3[(C - 1) * C + i] * y3b[i * CP + tid];
    sA[tid] = sigmoidf(d + adjrow[b * C + tid]);
  }
  __syncthreads();
  if (tid < L) {
    const float* xt = x + ((size_t)(b * S + t) * C) * L;
    float acc = 0.f;
    for (int j = 0; j < C; ++j) acc += sA[j] * xt[j * L + tid];
    vout[(size_t)b * LP + tid] = acc;
  }
}

// g (B x GCN) = V (B x LP) @ gcn_w_padded (LP x GCN) + gcn_b
__global__ void g_gemm_k(const float* __restrict__ v, const float* __restrict__ gwp,
                         const float* __restrict__ gb, float* __restrict__ g) {
  int n0 = blockIdx.x * 16, m0 = blockIdx.y * 16;
  int lane = (int)(threadIdx.x & 31);
  int arow = m0 + (lane & 15);
  int col = n0 + (lane & 15);
  v8f acc = V8F_ZERO;
  for (int k0 = 0; k0 < LP; k0 += 4) {
    int k = k0 + ((lane >> 4) << 1);
    v2f a = *(const v2f*)(v + (size_t)arow * LP + k);
    v2f bf;
    bf.x = gwp[(size_t)k * GCN + col];
    bf.y = gwp[(size_t)(k + 1) * GCN + col];
    acc = __builtin_amdgcn_wmma_f32_16x16x4_f32(false, a, false, bf, (short)0, acc, false, false);
  }
  int rbase = m0 + ((lane >> 4) << 3);
#pragma unroll
  for (int r = 0; r < 8; ++r)
    g[(size_t)(rbase + r) * GCN + col] = acc[r] + gb[col];
}

// GRU step: three 64x128x64 fp32 WMMA GEMMs in one block.
// cat=[g,h] staged in LDS; r*h folded into cat between phases -> no per-lane selects.
__global__ void gru_step_k(const float* __restrict__ g, const float* __restrict__ h_in,
                           float* __restrict__ h_out,
                           const float* __restrict__ w_r, const float* __restrict__ wb_r,
                           const float* __restrict__ b_r,
                           const float* __restrict__ w_u, const float* __restrict__ wb_u,
                           const float* __restrict__ b_u,
                           const float* __restrict__ w_c, const float* __restrict__ wb_c,
                           const float* __restrict__ b_c) {
  __shared__ float sCat[B * 128];
  __shared__ float sR[B * GRU];
  __shared__ float sU[B * GRU];
  int tid = (int)threadIdx.x;
  int wave = tid >> 5;
  int lane = tid & 31;
  for (int i = tid; i < B * GRU; i += 256) {
    int row = i >> 6, cc = i & 63;
    sCat[row * 128 + cc] = g[i];
    sCat[row * 128 + 64 + cc] = h_in[i];
  }
  __syncthreads();
  // phase 1: r and u gates (32 tile jobs)
  for (int job = wave; job < 32; job += 8) {
    int kind = job >> 4, tile = job & 15;
    int m0 = (tile >> 2) * 16, n0 = (tile & 3) * 16;
    const float* W = kind ? w_u : w_r;
    const float* bA = kind ? wb_u : wb_r;
    const float* bB = kind ? b_u : b_r;
    int arow = m0 + (lane & 15);
    int col = n0 + (lane & 15);
    v8f acc = V8F_ZERO;
    for (int k0 = 0; k0 < GCN + GRU; k0 += 4) {
      int k = k0 + ((lane >> 4) << 1);
      v2f a = *(const v2f*)(sCat + arow * 128 + k);
      v2f bf;
      bf.x = W[k * GRU + col];
      bf.y = W[(k + 1) * GRU + col];
      acc = __builtin_amdgcn_wmma_f32_16x16x4_f32(false, a, false, bf, (short)0, acc, false, false);
    }
    int rbase = m0 + ((lane >> 4) << 3);
    float bias = bA[col] + bB[col];
    float* Sm = kind ? sU : sR;
#pragma unroll
    for (int r = 0; r < 8; ++r)
      Sm[(rbase + r) * GRU + col] = sigmoidf(acc[r] + bias);
  }
  __syncthreads();
  // fold r*h into cat's h half
  for (int i = tid; i < B * GRU; i += 256) {
    int row = i >> 6, cc = i & 63;
    sCat[row * 128 + 64 + cc] *= sR[i];
  }
  __syncthreads();
  // phase 2: candidate + state update (16 tile jobs)
  for (int job = wave; job < 16; job += 8) {
    int m0 = (job >> 2) * 16, n0 = (job & 3) * 16;
    int arow = m0 + (lane & 15);
    int col = n0 + (lane & 15);
    v8f acc = V8F_ZERO;
    for (int k0 = 0; k0 < GCN + GRU; k0 += 4) {
      int k = k0 + ((lane >> 4) << 1);
      v2f a = *(const v2f*)(sCat + arow * 128 + k);
      v2f bf;
      bf.x = w_c[k * GRU + col];
      bf.y = w_c[(k + 1) * GRU + col];
      acc = __builtin_amdgcn_wmma_f32_16x16x4_f32(false, a, false, bf, (short)0, acc, false, false);
    }
    int rbase = m0 + ((lane >> 4) << 3);
    float bias = wb_c[col] + b_c[col];
#pragma unroll
    for (int r = 0; r < 8; ++r) {
      int row = rbase + r;
      float ct = tanhf(acc[r] + bias);
      float u = sU[row * GRU + col];
      h_out[row * GRU + col] = u * h_in[row * GRU + col] + (1.f - u) * ct;
    }
  }
}

// ---------------------------------------------------------------------------
// utility kernels
// ---------------------------------------------------------------------------
__global__ void fill_zero_k(float* __restrict__ p, int n) {
  int i = blockIdx.x * blockDim.x + threadIdx.x;
  if (i < n) p[i] = 0.f;
}

__global__ void pad_copy_k(const float* __restrict__ src, int sR, int sC,
                           float* __restrict__ dst, int dR, int dC) {
  int i = blockIdx.x * blockDim.x + threadIdx.x;
  if (i >= dR * dC) return;
  int r = i / dC, c = i % dC;
  dst[i] = (r < sR && c < sC) ? src[r * sC + c] : 0.f;
}

__global__ void copy_k(const float* __restrict__ src, float* __restrict__ dst, int n) {
  int i = blockIdx.x * blockDim.x + threadIdx.x;
  if (i < n) dst[i] = src[i];
}

// ---------------------------------------------------------------------------
// Host-side orchestration
// ---------------------------------------------------------------------------
extern "C" void kernel_launch(void* const* d_in, const int* in_sizes, int n_in,
                              void* d_out, int out_size, void* d_ws, size_t ws_size,
                              hipStream_t stream) {
  const float* x     = (const float*)d_in[0];
  const float* se_w1 = (const float*)d_in[1];
  const float* se_w2 = (const float*)d_in[2];
  const float* dw_k1 = (const float*)d_in[3];
  const float* dw_b1 = (const float*)d_in[4];
  const float* pw_k1 = (const float*)d_in[5];
  const float* pw_b1 = (const float*)d_in[6];
  const float* dw_k2 = (const float*)d_in[7];
  const float* dw_b2 = (const float*)d_in[8];
  const float* pw_k2 = (const float*)d_in[9];
  const float* pw_b2 = (const float*)d_in[10];
  const float* dw_k3 = (const float*)d_in[11];
  const float* dw_b3 = (const float*)d_in[12];
  const float* pw_k3 = (const float*)d_in[13];
  const float* pw_b3 = (const float*)d_in[14];
  const float* gcn_w = (const float*)d_in[15];
  const float* gcn_b = (const float*)d_in[16];
  const float* w_r   = (const float*)d_in[17];
  const float* wb_r  = (const float*)d_in[18];
  const float* w_u   = (const float*)d_in[19];
  const float* wb_u  = (const float*)d_in[20];
  const float* w_c   = (const float*)d_in[21];
  const float* wb_c  = (const float*)d_in[22];
  const float* b_r   = (const float*)d_in[23];
  const float* b_u   = (const float*)d_in[24];
  const float* b_c   = (const float*)d_in[25];
  float* out = (float*)d_out;
  (void)ws_size; (void)n_in; (void)in_sizes; (void)out_size;

  // workspace carve-up (floats)
  float* ws = (float*)d_ws;
  size_t off = 0;
  float* sq     = ws + off; off += (size_t)B * S;
  float* a_se   = ws + off; off += (size_t)B * S;
  float* msg    = ws + off; off += (size_t)B * CP * LP2;
  float* cosT   = ws + off; off += (size_t)LP2 * LP2;
  float* sinT   = ws + off; off += (size_t)LP2 * LP2;
  float* fre    = ws + off; off += (size_t)B * CP * LP2;
  float* fim    = ws + off; off += (size_t)B * CP * LP2;
  float* fab    = ws + off; off += (size_t)B * CP * LP2;
  float* arow   = ws + off; off += (size_t)B * C;
  float* y1     = ws + off; off += (size_t)B * CP * L1P;
  float* z1     = ws + off; off += (size_t)B * CP * L1P;
  float* y2     = ws + off; off += (size_t)B * CP * L2P;
  float* z2     = ws + off; off += (size_t)B * CP * L2P;
  float* y3     = ws + off; off += (size_t)B * CP * CP;
  float* vrow   = ws + off; off += (size_t)B * LP;
  float* gbuf   = ws + off; off += (size_t)B * GCN;
  float* hA     = ws + off; off += (size_t)B * GRU;
  float* hB     = ws + off; off += (size_t)B * GRU;
  float* pwk1p  = ws + off; off += (size_t)CP * CP;
  float* pwk2p  = ws + off; off += (size_t)CP * CP;
  float* pwb1p  = ws + off; off += (size_t)CP;
  float* pwb2p  = ws + off; off += (size_t)CP;
  float* gcnwp  = ws + off; off += (size_t)LP * GCN;

  // --- one-time (per call) prep: zero pads + padded weight copies
  fill_zero_k<<<(B * CP * LP2 + 255) / 256, 256, 0, stream>>>(msg, B * CP * LP2);
  fill_zero_k<<<(B * CP * L1P + 255) / 256, 256, 0, stream>>>(y1, B * CP * L1P);
  fill_zero_k<<<(B * CP * L2P + 255) / 256, 256, 0, stream>>>(y2, B * CP * L2P);
  fill_zero_k<<<(B * LP + 255) / 256, 256, 0, stream>>>(vrow, B * LP);
  pad_copy_k<<<(CP * CP + 255) / 256, 256, 0, stream>>>(pw_k1, C, C, pwk1p, CP, CP);
  pad_copy_k<<<(CP * CP + 255) / 256, 256, 0, stream>>>(pw_k2, C, C, pwk2p, CP, CP);
  pad_copy_k<<<(CP + 255) / 256, 256, 0, stream>>>(pw_b1, 1, C, pwb1p, 1, CP);
  pad_copy_k<<<(CP + 255) / 256, 256, 0, stream>>>(pw_b2, 1, C, pwb2p, 1, CP);
  pad_copy_k<<<(LP * GCN + 255) / 256, 256, 0, stream>>>(gcn_w, L, GCN, gcnwp, LP, GCN);

  // --- SE + max over segments
  se_mean_k<<<B * S, 256, 0, stream>>>(x, sq);
  se_mlp_k<<<B, 32, 0, stream>>>(sq, se_w1, se_w2, a_se);
  max_seg_k<<<(B * C * L + 255) / 256, 256, 0, stream>>>(x, a_se, msg);

  // --- iCOH: DFT (fp32 WMMA) then row-61 adjacency
  make_dft_k<<<(LP2 * LP2 + 255) / 256, 256, 0, stream>>>(cosT, sinT);
  dft_wmma_k<<<dim3(LP2 / 16, CP / 16, B), 32, 0, stream>>>(msg, cosT, sinT, fre, fim, fab);
  icoh_row_k<<<B, 64, 0, stream>>>(fre, fim, fab, arow);

  // --- recurrent scan; h double-buffered
  fill_zero_k<<<(B * GRU + 255) / 256, 256, 0, stream>>>(hA, B * GRU);
  for (int t = 0; t < S; ++t) {
    const float* xt0 = x + (size_t)t * C * L;
    dw_conv_k<<<dim3(C, B), 128, 64 * sizeof(float), stream>>>(
        xt0, (size_t)S * C * L, L, dw_k1, dw_b1, 64, y1, L1P, L1);
    pw_gemm_k<<<dim3(L1P / 16, 1, B), 128, 0, stream>>>(pwk1p, pwb1p, y1, z1, L1P);
    dw_conv_k<<<dim3(C, B), 96, 32 * sizeof(float), stream>>>(
        z1, (size_t)CP * L1P, L1P, dw_k2, dw_b2, 32, y2, L2P, L2);
    pw_gemm_k<<<dim3(L2P / 16, 1, B), 128, 0, stream>>>(pwk2p, pwb2p, y2, z2, L2P);
    dw_conv_k<<<dim3(C, B), 64, 16 * sizeof(float), stream>>>(
        z2, (size_t)CP * L2P, L2P, dw_k3, dw_b3, 14, y3, CP, L3);
    row_stage_k<<<B, 192, 0, stream>>>(x, t, pw_k3, pw_b3, y3, arow, vrow);
    g_gemm_k<<<dim3(GCN / 16, B / 16, 1), 32, 0, stream>>>(vrow, gcnwp, gcn_b, gbuf);
    const float* hin = (t & 1) ? hB : hA;
    float* hout      = (t & 1) ? hA : hB;
    gru_step_k<<<1, 256, 0, stream>>>(gbuf, hin, hout, w_r, wb_r, b_r,
                                      w_u, wb_u, b_u, w_c, wb_c, b_c);
  }
  // S=30 steps: final state lives in hA (t=29 wrote hA)
  copy_k<<<(B * GRU + 255) / 256, 256, 0, stream>>>(hA, out, B * GRU);
}